// CyberbullyingDetector_18554258719470
// MI455X (gfx1250) — compile-verified
//
#include <hip/hip_runtime.h>
#include <hip/hip_bf16.h>

typedef __attribute__((ext_vector_type(2))) float v2f;
typedef __attribute__((ext_vector_type(8))) float v8f;

#define N_NODES_C 100000
#define B_C 256

// ---------------------------------------------------------------------------
// Generic fp32 GEMM via V_WMMA_F32_16X16X4_F32.
// C[M,N] = act(A[M,K] @ W[K,N] + bias), M%16==0, K%4==0, N%16==0.
// One block = one 16-row strip of A (staged in LDS), wave w owns cols [16w,16w+16).
// blockDim.x must be 2*N (i.e. 32 * N/16 lanes).
// ---------------------------------------------------------------------------
__global__ void gemm16_wmma(const float* __restrict__ A, const float* __restrict__ W,
                            const float* __restrict__ bias, float* __restrict__ C,
                            int M, int K, int N, int act) {
    extern __shared__ float sA[];                 // 16*K floats
    const int tid = threadIdx.x;
    const int rowBlock = blockIdx.x;
    const float* Ablk = A + (size_t)rowBlock * 16 * K;
    const int total = 16 * K;
    for (int i = tid; i < total; i += blockDim.x) sA[i] = Ablk[i];
    __syncthreads();

    const int wave = tid >> 5;
    const int lane = tid & 31;
    const int colBase = wave * 16;
    const int col  = colBase + (lane & 15);
    const int rsel = lane >> 4;                   // 0: lanes 0-15, 1: lanes 16-31
    const int kb   = rsel * 2;                    // K sub-offset for A/B layout
    const int arow = lane & 15;

    v8f acc = {};
    for (int k0 = 0; k0 < K; k0 += 4) {
        v2f a, b;
        a.x = sA[arow * K + k0 + kb];
        a.y = sA[arow * K + k0 + kb + 1];
        b.x = W[(size_t)(k0 + kb) * N + col];
        b.y = W[(size_t)(k0 + kb + 1) * N + col];
        acc = __builtin_amdgcn_wmma_f32_16x16x4_f32(
                  false, a, false, b, (short)0, acc, false, false);
    }

    const float bcol = bias ? bias[col] : 0.0f;
#pragma unroll
    for (int r = 0; r < 8; ++r) {
        int row = rowBlock * 16 + r + rsel * 8;
        float v = acc[r] + bcol;
        if (act == 1) v = v > 0.0f ? v : 0.0f;    // relu
        C[(size_t)row * N + col] = v;
    }
}

// ---------------------------------------------------------------------------
// GCN helpers
// ---------------------------------------------------------------------------
__global__ void zero_f(float* p, int n) {
    int i = blockIdx.x * blockDim.x + threadIdx.x;
    for (; i < n; i += gridDim.x * blockDim.x) p[i] = 0.0f;
}

__global__ void degree_kernel(const int* __restrict__ dst, float* deg, int E) {
    int i = blockIdx.x * blockDim.x + threadIdx.x;
    for (; i < E; i += gridDim.x * blockDim.x) atomicAdd(&deg[dst[i]], 1.0f);
}

__global__ void rsqrt_kernel(float* dis, int n) {
    int i = blockIdx.x * blockDim.x + threadIdx.x;
    if (i < n) dis[i] = rsqrtf(dis[i] + 1.0f);
}

// out[n,c] = hw[n,c]*dis[n]^2 + bias[c]   (self-loop term seeds the accumulator)
__global__ void init_self(const float* __restrict__ hw, const float* __restrict__ dis,
                          const float* __restrict__ bias, float* __restrict__ out,
                          int Nn, int C) {
    int i = blockIdx.x * blockDim.x + threadIdx.x;
    int total = Nn * C;
    for (; i < total; i += gridDim.x * blockDim.x) {
        int n = i / C, c = i - n * C;
        float d = dis[n];
        out[i] = hw[i] * d * d + bias[c];
    }
}

// wave per edge: out[dst] += hw[src] * dis[src]*dis[dst]   (float4 lanes, f32 atomics)
__global__ void gcn_scatter(const float* __restrict__ hw, const int* __restrict__ src,
                            const int* __restrict__ dst, const float* __restrict__ dis,
                            float* out, int E, int C) {
    int gid  = blockIdx.x * blockDim.x + threadIdx.x;
    int edge = gid >> 5;
    int lane = gid & 31;
    if (edge >= E) return;
    int s = src[edge], d = dst[edge];
    float norm = dis[s] * dis[d];
    const float* hp = hw + (size_t)s * C;
    float* op = out + (size_t)d * C;
    for (int c = lane * 4; c < C; c += 128) {
        float4 m = *(const float4*)(hp + c);
        atomicAdd(op + c + 0, m.x * norm);
        atomicAdd(op + c + 1, m.y * norm);
        atomicAdd(op + c + 2, m.z * norm);
        atomicAdd(op + c + 3, m.w * norm);
    }
}

__global__ void elu_kernel(float* p, int n) {
    int i = blockIdx.x * blockDim.x + threadIdx.x;
    for (; i < n; i += gridDim.x * blockDim.x) {
        float x = p[i];
        p[i] = x > 0.0f ? x : (__expf(x) - 1.0f);
    }
}

// wave per node: pooled segment-sum + counts
__global__ void pool_kernel(const float* __restrict__ h, const int* __restrict__ batch,
                            float* sums, float* counts, int Nn) {
    int gid  = blockIdx.x * blockDim.x + threadIdx.x;
    int node = gid >> 5;
    int lane = gid & 31;
    if (node >= Nn) return;
    int b = batch[node];
    const float* hp = h + (size_t)node * 256;
    float* sp = sums + (size_t)b * 256;
    for (int c = lane * 4; c < 256; c += 128) {
        float4 v = *(const float4*)(hp + c);
        atomicAdd(sp + c + 0, v.x);
        atomicAdd(sp + c + 1, v.y);
        atomicAdd(sp + c + 2, v.z);
        atomicAdd(sp + c + 3, v.w);
    }
    if (lane == 0) atomicAdd(counts + b, 1.0f);
}

__global__ void pool_div(const float* sums, const float* counts, float* pooled) {
    int i = blockIdx.x * blockDim.x + threadIdx.x;
    if (i < B_C * 256) pooled[i] = sums[i] / counts[i >> 8];
}

__global__ void concat2(const float* __restrict__ a, const float* __restrict__ b,
                        float* __restrict__ o, int rows, int ca, int cb) {
    int i = blockIdx.x * blockDim.x + threadIdx.x;
    int cw = ca + cb, total = rows * cw;
    for (; i < total; i += gridDim.x * blockDim.x) {
        int r = i / cw, c = i - r * cw;
        o[i] = (c < ca) ? a[(size_t)r * ca + c] : b[(size_t)r * cb + (c - ca)];
    }
}

// two 2-class heads with log-softmax; out = [bullying(256x2) | aggression(256x2)]
__global__ void heads_kernel(const float* __restrict__ fin,
                             const float* __restrict__ Wb, const float* __restrict__ bb,
                             const float* __restrict__ Wa, const float* __restrict__ ba,
                             float* out) {
    int r = blockIdx.x * blockDim.x + threadIdx.x;
    if (r >= B_C) return;
    float z0 = bb[0], z1 = bb[1], y0 = ba[0], y1 = ba[1];
    const float* f = fin + (size_t)r * 512;
    for (int k = 0; k < 512; ++k) {
        float v = f[k];
        z0 += v * Wb[2 * k];     z1 += v * Wb[2 * k + 1];
        y0 += v * Wa[2 * k];     y1 += v * Wa[2 * k + 1];
    }
    float mz = fmaxf(z0, z1);
    float lz = mz + logf(__expf(z0 - mz) + __expf(z1 - mz));
    out[r * 2 + 0] = z0 - lz;
    out[r * 2 + 1] = z1 - lz;
    float my = fmaxf(y0, y1);
    float ly = my + logf(__expf(y0 - my) + __expf(y1 - my));
    out[2 * B_C + r * 2 + 0] = y0 - ly;
    out[2 * B_C + r * 2 + 1] = y1 - ly;
}

// ---------------------------------------------------------------------------
static inline int cdiv(int a, int b) { return (a + b - 1) / b; }

extern "C" void kernel_launch(void* const* d_in, const int* in_sizes, int n_in,
                              void* d_out, int out_size, void* d_ws, size_t ws_size,
                              hipStream_t stream) {
    const float* text  = (const float*)d_in[0];
    const float* video = (const float*)d_in[1];
    const float* x     = (const float*)d_in[2];
    const int*   eidx  = (const int*)d_in[3];
    const int*   batch = (const int*)d_in[4];
    const float *Wt = (const float*)d_in[5],  *bt = (const float*)d_in[6];
    const float *Wv = (const float*)d_in[7],  *bv = (const float*)d_in[8];
    // d_in[9..12] = Wq,bq,Wk,bk : provably dead (softmax over length-1 axis == 1)
    const float *Wva = (const float*)d_in[13], *bva = (const float*)d_in[14];
    const float *Wo  = (const float*)d_in[15], *bo  = (const float*)d_in[16];
    const float *Wf  = (const float*)d_in[17], *bf  = (const float*)d_in[18];
    const float *G1W = (const float*)d_in[19], *G1b = (const float*)d_in[20];
    const float *G2W = (const float*)d_in[21], *G2b = (const float*)d_in[22];
    const float *G3W = (const float*)d_in[23], *G3b = (const float*)d_in[24];
    const float *Wff = (const float*)d_in[25], *bff = (const float*)d_in[26];
    const float *Wb  = (const float*)d_in[27], *bb  = (const float*)d_in[28];
    const float *Wa  = (const float*)d_in[29], *ba  = (const float*)d_in[30];

    const int E  = in_sizes[3] / 2;
    const int NN = in_sizes[4];              // 100000
    const int* esrc = eidx;
    const int* edst = eidx + E;

    // workspace carve-up (floats)
    float* ws = (float*)d_ws;
    size_t off = 0;
    float* bufA     = ws + off; off += (size_t)NN * 256;   // hw scratch (max 256 ch)
    float* bufB     = ws + off; off += (size_t)NN * 256;   // accum / h scratch
    float* dis      = ws + off; off += NN;                 // degree -> rsqrt
    float* text_enc = ws + off; off += B_C * 512;
    float* vid_enc  = ws + off; off += B_C * 512;
    float* attn     = ws + off; off += B_C * 512;
    float* text_att = ws + off; off += B_C * 512;
    float* catTV    = ws + off; off += B_C * 1024;
    float* fusedb   = ws + off; off += B_C * 512;
    float* sums     = ws + off; off += B_C * 256;
    float* counts   = ws + off; off += B_C;
    float* pooled   = ws + off; off += B_C * 256;
    float* catFP    = ws + off; off += B_C * 768;
    float* finalb   = ws + off; off += B_C * 512;

    // ---------------- dense front end ----------------
    // text_enc = relu(text @ Wt + bt)            [256,768]x[768,512]
    gemm16_wmma<<<B_C / 16, 1024, 16 * 768 * 4, stream>>>(text, Wt, bt, text_enc, B_C, 768, 512, 1);
    // video_enc = relu(video @ Wv + bv)          [256,512]x[512,512]
    gemm16_wmma<<<B_C / 16, 1024, 16 * 512 * 4, stream>>>(video, Wv, bv, vid_enc, B_C, 512, 512, 1);
    // attn == v (softmax over 1 element is identity): attn = video_enc @ Wva + bva
    gemm16_wmma<<<B_C / 16, 1024, 16 * 512 * 4, stream>>>(vid_enc, Wva, bva, attn, B_C, 512, 512, 0);
    // text_attended = attn @ Wo + bo
    gemm16_wmma<<<B_C / 16, 1024, 16 * 512 * 4, stream>>>(attn, Wo, bo, text_att, B_C, 512, 512, 0);
    // fused = relu([text_att | video_enc] @ Wf + bf)
    concat2<<<cdiv(B_C * 1024, 256), 256, 0, stream>>>(text_att, vid_enc, catTV, B_C, 512, 512);
    gemm16_wmma<<<B_C / 16, 1024, 16 * 1024 * 4, stream>>>(catTV, Wf, bf, fusedb, B_C, 1024, 512, 1);

    // ---------------- GCN stack ----------------
    // dis = rsqrt(indeg + 1)
    zero_f<<<cdiv(NN, 256), 256, 0, stream>>>(dis, NN);
    degree_kernel<<<cdiv(E, 256), 256, 0, stream>>>(edst, dis, E);
    rsqrt_kernel<<<cdiv(NN, 256), 256, 0, stream>>>(dis, NN);

    const int scatterBlocks = cdiv(E * 32, 256);

    // layer 1: 64 -> 128, elu
    gemm16_wmma<<<NN / 16, 256, 16 * 64 * 4, stream>>>(x, G1W, nullptr, bufA, NN, 64, 128, 0);
    init_self<<<cdiv(NN * 128, 256), 256, 0, stream>>>(bufA, dis, G1b, bufB, NN, 128);
    gcn_scatter<<<scatterBlocks, 256, 0, stream>>>(bufA, esrc, edst, dis, bufB, E, 128);
    elu_kernel<<<cdiv(NN * 128, 256), 256, 0, stream>>>(bufB, NN * 128);

    // layer 2: 128 -> 128, elu
    gemm16_wmma<<<NN / 16, 256, 16 * 128 * 4, stream>>>(bufB, G2W, nullptr, bufA, NN, 128, 128, 0);
    init_self<<<cdiv(NN * 128, 256), 256, 0, stream>>>(bufA, dis, G2b, bufB, NN, 128);
    gcn_scatter<<<scatterBlocks, 256, 0, stream>>>(bufA, esrc, edst, dis, bufB, E, 128);
    elu_kernel<<<cdiv(NN * 128, 256), 256, 0, stream>>>(bufB, NN * 128);

    // layer 3: 128 -> 256, linear
    gemm16_wmma<<<NN / 16, 512, 16 * 128 * 4, stream>>>(bufB, G3W, nullptr, bufA, NN, 128, 256, 0);
    init_self<<<cdiv(NN * 256, 256), 256, 0, stream>>>(bufA, dis, G3b, bufB, NN, 256);
    gcn_scatter<<<scatterBlocks, 256, 0, stream>>>(bufA, esrc, edst, dis, bufB, E, 256);

    // per-graph mean pooling
    zero_f<<<cdiv(B_C * 256 + B_C, 256), 256, 0, stream>>>(sums, B_C * 256 + B_C); // sums+counts contiguous
    pool_kernel<<<cdiv(NN * 32, 256), 256, 0, stream>>>(bufB, batch, sums, counts, NN);
    pool_div<<<cdiv(B_C * 256, 256), 256, 0, stream>>>(sums, counts, pooled);

    // ---------------- dense back end ----------------
    concat2<<<cdiv(B_C * 768, 256), 256, 0, stream>>>(fusedb, pooled, catFP, B_C, 512, 256);
    gemm16_wmma<<<B_C / 16, 1024, 16 * 768 * 4, stream>>>(catFP, Wff, bff, finalb, B_C, 768, 512, 1);
    heads_kernel<<<cdiv(B_C, 256), 256, 0, stream>>>(finalb, Wb, bb, Wa, ba, (float*)d_out);
}